// weight_loss2_833223655948
// MI455X (gfx1250) — compile-verified
//
#include <hip/hip_runtime.h>

// Vector types
typedef float f4  __attribute__((ext_vector_type(4)));
typedef float v2f __attribute__((ext_vector_type(2)));
typedef float v8f __attribute__((ext_vector_type(8)));

// Branchless bucket weights:
// [0,20)->1, [20,30)->2, [30,40)->5, [40,60)->10, [60,inf)->30, (<0 -> 0)
__device__ __forceinline__ float bucket_w(float l) {
    float w = 0.0f;
    w = (l >= 0.0f)  ? 1.0f  : w;
    w = (l >= 20.0f) ? 2.0f  : w;
    w = (l >= 30.0f) ? 5.0f  : w;
    w = (l >= 40.0f) ? 10.0f : w;
    w = (l >= 60.0f) ? 30.0f : w;
    return w;
}

// Wave32 all-lane sum using V_WMMA_F32_16X16X4_F32.
// A: VGPR0 = p (per lane), VGPR1 = 0  =>  A[m][0]=p_m (lanes 0-15), A[m][2]=p_{m+16}
// B: all ones (4x16)                  =>  D[m][n] = p_m + p_{m+16}
// Per-lane sum of the 8 D VGPRs gives sum over half the M rows; one cross-half
// shuffle-add completes the 32-lane reduction (result in every lane).
__device__ __forceinline__ float wave_sum_wmma(float p) {
    v2f a; a[0] = p;    a[1] = 0.0f;
    v2f b; b[0] = 1.0f; b[1] = 1.0f;
    v8f c = {};
    v8f d = __builtin_amdgcn_wmma_f32_16x16x4_f32(
        /*neg_a=*/false, a, /*neg_b=*/false, b,
        /*c_mod=*/(short)0, c, /*reuse_a=*/false, /*reuse_b=*/false);
    float s = ((d[0] + d[1]) + (d[2] + d[3])) + ((d[4] + d[5]) + (d[6] + d[7]));
    s += __shfl_xor(s, 16, 32);
    return s;
}

// Block reduction: WMMA wave sums -> LDS -> thread 0 accumulates (fixed order,
// deterministic). Valid result only in thread 0.
__device__ __forceinline__ float block_sum(float p) {
    __shared__ float sm[32];
    float s = wave_sum_wmma(p);           // every thread reaches this (EXEC all 1s)
    const int wave = threadIdx.x >> 5;
    const int lane = threadIdx.x & 31;
    if (lane == 0) sm[wave] = s;
    __syncthreads();
    float tot = 0.0f;
    if (threadIdx.x == 0) {
        const int nw = blockDim.x >> 5;
        for (int i = 0; i < nw; ++i) tot += sm[i];
    }
    return tot;
}

__global__ void wl2_partial(const float* __restrict__ label,
                            const float* __restrict__ predict,
                            float* __restrict__ partials, int n) {
    const int  n4     = n >> 2;
    const long stride = (long)gridDim.x * blockDim.x;
    const long tid    = (long)blockIdx.x * blockDim.x + threadIdx.x;

    const f4* l4 = (const f4*)label;
    const f4* p4 = (const f4*)predict;

    float acc = 0.0f;
    for (long i = tid; i < n4; i += stride) {
        f4 l = __builtin_nontemporal_load(&l4[i]);   // global_load_b128, TH=NT
        f4 p = __builtin_nontemporal_load(&p4[i]);
        #pragma unroll
        for (int j = 0; j < 4; ++j) {
            float w  = bucket_w(l[j]);
            float df = l[j] - p[j];
            acc = fmaf(w * df, df, acc);
        }
    }
    // Scalar tail (n not divisible by 4): first `tail` threads take one each.
    const int tail = n & 3;
    if (tid < tail) {
        const int i = (n4 << 2) + (int)tid;
        float lv = label[i], pv = predict[i];
        float w  = bucket_w(lv);
        float df = lv - pv;
        acc = fmaf(w * df, df, acc);
    }
    // Divergence above has reconverged; all lanes active for WMMA.
    float tot = block_sum(acc);
    if (threadIdx.x == 0) partials[blockIdx.x] = tot;
}

__global__ void wl2_final(const float* __restrict__ partials, int nparts,
                          float* __restrict__ out, float inv_n) {
    float acc = 0.0f;
    for (int i = threadIdx.x; i < nparts; i += blockDim.x) acc += partials[i];
    float tot = block_sum(acc);
    if (threadIdx.x == 0) out[0] = tot * inv_n;
}

extern "C" void kernel_launch(void* const* d_in, const int* in_sizes, int n_in,
                              void* d_out, int out_size, void* d_ws, size_t ws_size,
                              hipStream_t stream) {
    const float* label   = (const float*)d_in[0];
    const float* predict = (const float*)d_in[1];
    float*       out     = (float*)d_out;
    float*       parts   = (float*)d_ws;
    const int    n       = in_sizes[0];

    // 1024 partials (4 KiB of d_ws); clamp if workspace is unexpectedly small.
    int nparts = (int)(ws_size / sizeof(float));
    if (nparts > 1024) nparts = 1024;
    if (nparts < 1)    nparts = 1;

    // Pass 1: 1024 blocks x 256 threads (8 waves/block) streaming reduction.
    wl2_partial<<<nparts, 256, 0, stream>>>(label, predict, parts, n);
    // Pass 2: single block, deterministic final combine + scale by 1/n.
    wl2_final<<<1, 256, 0, stream>>>(parts, nparts, out, 1.0f / (float)n);
}